// EglGATConvTest_19774029430811
// MI455X (gfx1250) — compile-verified
//
#include <hip/hip_runtime.h>
#include <hip/hip_bf16.h>

typedef __attribute__((ext_vector_type(2))) float v2f;
typedef __attribute__((ext_vector_type(8))) float v8f;

#define N_NODES   100000
#define E_EDGES   1600000
#define IN_F      256
#define HD        128      // H * D
#define NHEADS    4
#define DOUT      32
#define NEG_SLOPE 0.2f
#define LDS_STRIDE 260     // 256 + 4 pad: conflict-free f32x2 reads, 16B-aligned rows

// ---------------------------------------------------------------------------
// 0) zero the accumulation targets (agg half of d_out, softmax denominators).
//    Must run every call: harness replays the graph without re-poisoning.
// ---------------------------------------------------------------------------
__global__ void zero_kernel(float4* __restrict__ agg4, float4* __restrict__ denom4) {
    long long i = (long long)blockIdx.x * blockDim.x + threadIdx.x;
    const long long n4 = (long long)N_NODES * HD / 4;   // 3,200,000
    if (i < n4)       agg4[i]   = make_float4(0.f, 0.f, 0.f, 0.f);
    if (i < N_NODES)  denom4[i] = make_float4(0.f, 0.f, 0.f, 0.f);   // N*H floats
}

// ---------------------------------------------------------------------------
// 1) fused dual GEMM: ft = feat @ fc_w^T, res = feat @ res_w^T
//    WMMA f32 16x16x4. Block = 8 waves, one 16-node M-tile / block,
//    wave w owns column tile [16w, 16w+16). feat tile staged in LDS once.
// ---------------------------------------------------------------------------
__global__ __launch_bounds__(256) void proj_kernel(
        const float* __restrict__ feat,
        const float* __restrict__ fc_w,
        const float* __restrict__ res_w,
        float* __restrict__ ft,          // [N, HD] workspace
        float* __restrict__ rst)         // d_out[0:N*HD], receives res
{
    __shared__ __align__(16) float sfeat[16 * LDS_STRIDE];

    const int tid  = threadIdx.x;
    const int wave = tid >> 5;
    const int lane = tid & 31;
    const int half = lane >> 4;          // 0: lanes 0-15, 1: lanes 16-31
    const int mn   = lane & 15;          // row (A) / col (B) within tile
    const long long row0 = (long long)blockIdx.x * 16;

    // stage 16x256 feat tile: 1024 float4 slots, 4 per thread
    const float4* gfeat = (const float4*)(feat + row0 * IN_F);
#pragma unroll
    for (int i = 0; i < 4; ++i) {
        int s = tid + i * 256;           // float4 slot
        int r = s >> 6;                  // row 0..15
        int c = (s & 63) << 2;           // col 0..252
        *(float4*)&sfeat[r * LDS_STRIDE + c] = gfeat[s];
    }
    __syncthreads();

    const int c0 = wave * 16;
    v8f acc_fc  = {};
    v8f acc_res = {};
    const float* wrow_fc  = fc_w  + (long long)(c0 + mn) * IN_F;  // B[k][n]=w[n][k]
    const float* wrow_res = res_w + (long long)(c0 + mn) * IN_F;
    const float* arow     = &sfeat[mn * LDS_STRIDE];

#pragma unroll 4
    for (int kk = 0; kk < IN_F; kk += 4) {
        const int kc = kk + 2 * half;    // K pair owned by this half-wave
        v2f a  = *(const v2f*)&arow[kc];       // A: 16x4 f32 layout
        v2f bf = *(const v2f*)&wrow_fc[kc];    // B: 4x16 f32 layout
        v2f br = *(const v2f*)&wrow_res[kc];
        acc_fc  = __builtin_amdgcn_wmma_f32_16x16x4_f32(
                      false, a, false, bf, (short)0, acc_fc,  false, false);
        acc_res = __builtin_amdgcn_wmma_f32_16x16x4_f32(
                      false, a, false, br, (short)0, acc_res, false, false);
    }

    // D layout: VGPR r -> M = r + 8*half, N = lane&15
#pragma unroll
    for (int r = 0; r < 8; ++r) {
        long long row = row0 + r + 8 * half;
        ft [row * HD + c0 + mn] = acc_fc[r];
        rst[row * HD + c0 + mn] = acc_res[r];   // res staged into rst region
    }
}

// ---------------------------------------------------------------------------
// 2) per-node attention logits: el[n,h] = <ft[n,h,:], attn_l[h,:]>, same er
// ---------------------------------------------------------------------------
__global__ void eler_kernel(const float* __restrict__ ft,
                            const float* __restrict__ attn_l,
                            const float* __restrict__ attn_r,
                            float* __restrict__ el, float* __restrict__ er) {
    int t = blockIdx.x * blockDim.x + threadIdx.x;
    if (t >= N_NODES * NHEADS) return;
    int n = t >> 2, h = t & 3;
    const float* f  = ft + (long long)n * HD + h * DOUT;
    const float* al = attn_l + h * DOUT;
    const float* ar = attn_r + h * DOUT;
    float sl = 0.f, sr = 0.f;
#pragma unroll
    for (int d = 0; d < DOUT; ++d) {
        float v = f[d];
        sl += v * al[d];
        sr += v * ar[d];
    }
    el[t] = sl;
    er[t] = sr;
}

// ---------------------------------------------------------------------------
// 3) per-edge: ex = exp(leaky(el[src]+er[dst])); denom[dst] += ex (atomic)
// ---------------------------------------------------------------------------
__global__ void edge_kernel(const int* __restrict__ src, const int* __restrict__ dst,
                            const float* __restrict__ el, const float* __restrict__ er,
                            float* __restrict__ exb, float* __restrict__ denom) {
    long long t = (long long)blockIdx.x * blockDim.x + threadIdx.x;
    if (t >= (long long)E_EDGES * NHEADS) return;
    int e = (int)(t >> 2), h = (int)(t & 3);
    int s = src[e], d = dst[e];
    float v = el[s * NHEADS + h] + er[d * NHEADS + h];
    v = v > 0.f ? v : NEG_SLOPE * v;
    float x = expf(v);
    exb[t] = x;
    atomicAdd(&denom[d * NHEADS + h], x);
}

// ---------------------------------------------------------------------------
// 4) one wave per edge: agg[dst,:] += ft[src,:] * alpha   (128 cols, f32x4/lane)
// ---------------------------------------------------------------------------
__global__ __launch_bounds__(256) void scatter_kernel(
        const int* __restrict__ src, const int* __restrict__ dst,
        const float* __restrict__ exb, const float* __restrict__ denom,
        const float* __restrict__ ft, float* __restrict__ agg) {
    int wid  = (int)(((long long)blockIdx.x * blockDim.x + threadIdx.x) >> 5);
    int lane = threadIdx.x & 31;
    if (wid >= E_EDGES) return;
    int s = src[wid], d = dst[wid];
    int h = lane >> 3;                               // 4 lanes-of-8 per head
    float a = exb[wid * NHEADS + h] / denom[d * NHEADS + h];
    float4 f = *(const float4*)&ft[(long long)s * HD + lane * 4];
    float* base = agg + (long long)d * HD + lane * 4;
    atomicAdd(base + 0, f.x * a);
    atomicAdd(base + 1, f.y * a);
    atomicAdd(base + 2, f.z * a);
    atomicAdd(base + 3, f.w * a);
}

// ---------------------------------------------------------------------------
// 5) rst = res(+already stored) + agg
// ---------------------------------------------------------------------------
__global__ void finalize_kernel(float4* __restrict__ rst4, const float4* __restrict__ agg4) {
    long long i = (long long)blockIdx.x * blockDim.x + threadIdx.x;
    if (i >= (long long)N_NODES * HD / 4) return;
    float4 r = rst4[i];
    float4 g = agg4[i];
    r.x += g.x; r.y += g.y; r.z += g.z; r.w += g.w;
    rst4[i] = r;
}

// ---------------------------------------------------------------------------
extern "C" void kernel_launch(void* const* d_in, const int* in_sizes, int n_in,
                              void* d_out, int out_size, void* d_ws, size_t ws_size,
                              hipStream_t stream) {
    const float* feat   = (const float*)d_in[0];
    const int*   src    = (const int*)  d_in[1];
    const int*   dst    = (const int*)  d_in[2];
    const float* fc_w   = (const float*)d_in[3];
    const float* attn_l = (const float*)d_in[4];
    const float* attn_r = (const float*)d_in[5];
    const float* res_w  = (const float*)d_in[6];

    float* rst = (float*)d_out;                       // [N, HD]
    float* agg = rst + (long long)N_NODES * HD;       // [N, HD]

    float* ft    = (float*)d_ws;                      // N*HD
    float* exb   = ft  + (long long)N_NODES * HD;     // E*NHEADS
    float* el    = exb + (long long)E_EDGES * NHEADS; // N*NHEADS
    float* er    = el  + N_NODES * NHEADS;            // N*NHEADS
    float* denom = er  + N_NODES * NHEADS;            // N*NHEADS

    zero_kernel<<<(N_NODES * HD / 4 + 255) / 256, 256, 0, stream>>>(
        (float4*)agg, (float4*)denom);
    proj_kernel<<<N_NODES / 16, 256, 0, stream>>>(feat, fc_w, res_w, ft, rst);
    eler_kernel<<<(N_NODES * NHEADS + 255) / 256, 256, 0, stream>>>(
        ft, attn_l, attn_r, el, er);
    edge_kernel<<<(int)(((long long)E_EDGES * NHEADS + 255) / 256), 256, 0, stream>>>(
        src, dst, el, er, exb, denom);
    scatter_kernel<<<E_EDGES / 8, 256, 0, stream>>>(src, dst, exb, denom, ft, agg);
    finalize_kernel<<<(N_NODES * HD / 4 + 255) / 256, 256, 0, stream>>>(
        (float4*)rst, (const float4*)agg);
}